// GIN_4604204941844
// MI455X (gfx1250) — compile-verified
//
#include <hip/hip_runtime.h>
#include <hip/hip_bf16.h>

#define NNODES 50000
#define NPAD   50016   // multiple of 32 (WMMA block rows); pad rows stay finite & ignored
#define NEDGES 800000
#define CIN    128
#define HDIM   64
#define OUTDIM 32
#define BGRAPH 512
#define BN_EPS 1e-5f

typedef float v2f __attribute__((ext_vector_type(2)));
typedef float v8f __attribute__((ext_vector_type(8)));

enum { PRO_NONE = 0, PRO_BNRELU = 1 };
enum { EPI_NONE = 0, EPI_RELU = 1 };

// ---------------- utility kernels ----------------

__global__ void fill0_kernel(float* __restrict__ p, long long n) {
  long long i = (long long)blockIdx.x * blockDim.x + threadIdx.x;
  if (i < n) p[i] = 0.0f;
}

__global__ void add_rows_kernel(float* __restrict__ h, const float* __restrict__ x, long long n) {
  long long i = (long long)blockIdx.x * blockDim.x + threadIdx.x;
  if (i < n) h[i] += x[i];
}

// agg[dst[e]][c] += X[src[e]][c]; one thread per (edge, feature)
template <int C, int LOGC>
__global__ void scatter_add_kernel(const float* __restrict__ X,
                                   const long long* __restrict__ src,
                                   const long long* __restrict__ dst,
                                   float* __restrict__ agg) {
  long long t = (long long)blockIdx.x * blockDim.x + threadIdx.x;
  if (t >= (long long)NEDGES * C) return;
  const int e = (int)(t >> LOGC);
  const int c = (int)(t & (C - 1));
  const int s = (int)src[e];
  const int d = (int)dst[e];
  atomicAdd(&agg[(long long)d * C + c], X[(long long)s * C + c]);
}

// ---------------- WMMA GEMM: Out[M,NC] = f(A[M,K]) @ W[K,NC] + bias ----------------
// fp32 WMMA 16x16x4. One wave per 16x16 tile; 8 waves per block.
// A-frag (16x4 f32, ISA layout): lane<16 holds {A[r][k+0],A[r][k+1]}, lane>=16 {A[r][k+2],A[r][k+3]}
// B-frag (4x16 f32): lane<16 holds {W[k+0][n],W[k+1][n]}, lane>=16 {W[k+2][n],W[k+3][n]}
// C/D: VGPR r = row base+r (lanes 0-15) / base+8+r (lanes 16-31), col = lane&15
template <int K, int NC, int PRO, int EPI>
__global__ __launch_bounds__(256) void gemm_wmma_kernel(
    const float* __restrict__ A, const float* __restrict__ W,
    const float* __restrict__ bias,
    const float* __restrict__ bnscale, const float* __restrict__ bnshift,
    float* __restrict__ Out) {
  constexpr int CT = NC / 16;   // column tiles
  constexpr int RT = 8 / CT;    // row tiles per block (8 waves)
  const int wave = threadIdx.x >> 5;
  const int lane = threadIdx.x & 31;
  const int ct = wave % CT;
  const int rt = wave / CT;
  const int rowBase = (blockIdx.x * RT + rt) * 16;
  const int colBase = ct * 16;

  const int arow = rowBase + (lane & 15);
  const int koff = (lane >> 4) * 2;            // 0 for lanes 0-15, 2 for lanes 16-31
  const int bcol = colBase + (lane & 15);

  const float* __restrict__ aptr = A + (long long)arow * K + koff;

  v8f acc = {};
#pragma unroll
  for (int kk = 0; kk < K; kk += 4) {
    v2f a = *(const v2f*)(aptr + kk);          // 8B aligned (kk+koff even)
    if (PRO == PRO_BNRELU) {
      a.x = fmaxf(fmaf(a.x, bnscale[kk + koff],     bnshift[kk + koff]),     0.0f);
      a.y = fmaxf(fmaf(a.y, bnscale[kk + koff + 1], bnshift[kk + koff + 1]), 0.0f);
    }
    v2f b;
    b.x = W[(kk + koff) * NC + bcol];
    b.y = W[(kk + koff + 1) * NC + bcol];
    acc = __builtin_amdgcn_wmma_f32_16x16x4_f32(false, a, false, b, (short)0, acc,
                                                false, false);
  }

  const int srow = rowBase + (lane >> 4) * 8;
  const int scol = colBase + (lane & 15);
  const float bv = bias[scol];
#pragma unroll
  for (int r = 0; r < 8; ++r) {
    float v = acc[r] + bv;
    if (EPI == EPI_RELU) v = fmaxf(v, 0.0f);
    Out[(long long)(srow + r) * NC + scol] = v;
  }
}

// ---------------- BatchNorm training statistics (deterministic 2-pass) ----------------

__global__ void bn_partials_kernel(const float* __restrict__ pre,
                                   float* __restrict__ psum, float* __restrict__ psq) {
  __shared__ float ssum[4][HDIM];
  __shared__ float ssq[4][HDIM];
  const int c = threadIdx.x & 63;
  const int g = threadIdx.x >> 6;              // 0..3
  const int rowBeg = blockIdx.x * 256;
  int rowEnd = rowBeg + 256;
  if (rowEnd > NNODES) rowEnd = NNODES;
  float s = 0.0f, q = 0.0f;
  for (int r = rowBeg + g; r < rowEnd; r += 4) {
    const float v = pre[(long long)r * HDIM + c];
    s += v;
    q += v * v;
  }
  ssum[g][c] = s;
  ssq[g][c] = q;
  __syncthreads();
  if (g == 0) {
    psum[blockIdx.x * HDIM + c] = ssum[0][c] + ssum[1][c] + ssum[2][c] + ssum[3][c];
    psq[blockIdx.x * HDIM + c]  = ssq[0][c]  + ssq[1][c]  + ssq[2][c]  + ssq[3][c];
  }
}

__global__ void bn_finalize_kernel(const float* __restrict__ psum, const float* __restrict__ psq,
                                   int nblocks, const float* __restrict__ g,
                                   const float* __restrict__ bt,
                                   float* __restrict__ scale, float* __restrict__ shift) {
  const int c = threadIdx.x;                   // 64 threads
  float s = 0.0f, q = 0.0f;
  for (int b = 0; b < nblocks; ++b) {
    s += psum[b * HDIM + c];
    q += psq[b * HDIM + c];
  }
  const float inv_n = 1.0f / (float)NNODES;
  const float mean = s * inv_n;
  const float var = q * inv_n - mean * mean;   // biased variance (training mode)
  const float sc = g[c] * rsqrtf(var + BN_EPS);
  scale[c] = sc;
  shift[c] = fmaf(-mean, sc, bt[c]);
}

// ---------------- global_add_pool ----------------

__global__ void pool_kernel(const float* __restrict__ xf, const long long* __restrict__ batch,
                            float* __restrict__ pooled) {
  long long t = (long long)blockIdx.x * blockDim.x + threadIdx.x;
  if (t >= (long long)NNODES * HDIM) return;
  const int r = (int)(t >> 6);
  const int c = (int)(t & 63);
  atomicAdd(&pooled[(int)batch[r] * HDIM + c], xf[t]);
}

// ---------------- launcher ----------------

extern "C" void kernel_launch(void* const* d_in, const int* in_sizes, int n_in,
                              void* d_out, int out_size, void* d_ws, size_t ws_size,
                              hipStream_t stream) {
  const float* x = (const float*)d_in[0];
  const long long* eidx = (const long long*)d_in[1];
  const long long* src = eidx;               // edge_index[0]
  const long long* dst = eidx + NEDGES;      // edge_index[1]
  const long long* batch = (const long long*)d_in[2];
  // d_in[3] = batch_size scalar (known: 512)

  const float *cW1[3], *cb1[3], *cg1[3], *cbt1[3], *cW2[3], *cb2[3];
  for (int i = 0; i < 3; ++i) {
    const int base = 4 + i * 6;
    cW1[i]  = (const float*)d_in[base + 0];
    cb1[i]  = (const float*)d_in[base + 1];
    cg1[i]  = (const float*)d_in[base + 2];
    cbt1[i] = (const float*)d_in[base + 3];
    cW2[i]  = (const float*)d_in[base + 4];
    cb2[i]  = (const float*)d_in[base + 5];
  }
  const float* fW1 = (const float*)d_in[22];
  const float* fb1 = (const float*)d_in[23];
  const float* fW2 = (const float*)d_in[24];
  const float* fb2 = (const float*)d_in[25];

  // workspace carve-up (floats)
  float* ws = (float*)d_ws;
  float* buf0 = ws;   ws += (long long)NPAD * CIN;    // agg/h, layer 0 (25.6 MB)
  float* bufA = ws;   ws += (long long)NPAD * HDIM;   // pre-activation
  float* bufB = ws;   ws += (long long)NPAD * HDIM;   // layer output
  float* bufC = ws;   ws += (long long)NPAD * HDIM;   // agg/h, layers 1-2
  const int SB = (NNODES + 255) / 256;                // 196 stat blocks
  float* psum = ws;   ws += (long long)SB * HDIM;
  float* psq  = ws;   ws += (long long)SB * HDIM;
  float* bnsc = ws;   ws += HDIM;
  float* bnsh = ws;   ws += HDIM;
  float* pooled = ws; ws += (long long)BGRAPH * HDIM;
  float* hfin = ws;   ws += (long long)BGRAPH * HDIM;

  auto nblk = [](long long n, int bs) { return (unsigned)((n + bs - 1) / bs); };

  // ---- Layer 0 (cin=128) ----
  {
    const long long nAgg = (long long)NPAD * CIN;
    fill0_kernel<<<nblk(nAgg, 256), 256, 0, stream>>>(buf0, nAgg);
    scatter_add_kernel<128, 7>
        <<<nblk((long long)NEDGES * 128, 256), 256, 0, stream>>>(x, src, dst, buf0);
    add_rows_kernel<<<nblk((long long)NNODES * CIN, 256), 256, 0, stream>>>(
        buf0, x, (long long)NNODES * CIN);
    gemm_wmma_kernel<CIN, HDIM, PRO_NONE, EPI_NONE>
        <<<NPAD / 32, 256, 0, stream>>>(buf0, cW1[0], cb1[0], nullptr, nullptr, bufA);
    bn_partials_kernel<<<SB, 256, 0, stream>>>(bufA, psum, psq);
    bn_finalize_kernel<<<1, 64, 0, stream>>>(psum, psq, SB, cg1[0], cbt1[0], bnsc, bnsh);
    gemm_wmma_kernel<HDIM, HDIM, PRO_BNRELU, EPI_RELU>
        <<<NPAD / 32, 256, 0, stream>>>(bufA, cW2[0], cb2[0], bnsc, bnsh, bufB);
  }

  // ---- Layers 1, 2 (cin=64) ----
  for (int l = 1; l <= 2; ++l) {
    const long long nAgg = (long long)NPAD * HDIM;
    fill0_kernel<<<nblk(nAgg, 256), 256, 0, stream>>>(bufC, nAgg);
    scatter_add_kernel<64, 6>
        <<<nblk((long long)NEDGES * 64, 256), 256, 0, stream>>>(bufB, src, dst, bufC);
    add_rows_kernel<<<nblk((long long)NNODES * HDIM, 256), 256, 0, stream>>>(
        bufC, bufB, (long long)NNODES * HDIM);
    gemm_wmma_kernel<HDIM, HDIM, PRO_NONE, EPI_NONE>
        <<<NPAD / 32, 256, 0, stream>>>(bufC, cW1[l], cb1[l], nullptr, nullptr, bufA);
    bn_partials_kernel<<<SB, 256, 0, stream>>>(bufA, psum, psq);
    bn_finalize_kernel<<<1, 64, 0, stream>>>(psum, psq, SB, cg1[l], cbt1[l], bnsc, bnsh);
    gemm_wmma_kernel<HDIM, HDIM, PRO_BNRELU, EPI_RELU>
        <<<NPAD / 32, 256, 0, stream>>>(bufA, cW2[l], cb2[l], bnsc, bnsh, bufB);
  }

  // ---- global_add_pool + final MLP ----
  {
    const long long nPool = (long long)BGRAPH * HDIM;
    fill0_kernel<<<nblk(nPool, 256), 256, 0, stream>>>(pooled, nPool);
    pool_kernel<<<nblk((long long)NNODES * HDIM, 256), 256, 0, stream>>>(bufB, batch, pooled);
    gemm_wmma_kernel<HDIM, HDIM, PRO_NONE, EPI_RELU>
        <<<BGRAPH / 32, 256, 0, stream>>>(pooled, fW1, fb1, nullptr, nullptr, hfin);
    gemm_wmma_kernel<HDIM, OUTDIM, PRO_NONE, EPI_NONE>
        <<<BGRAPH / 64, 256, 0, stream>>>(hfin, fW2, fb2, nullptr, nullptr, (float*)d_out);
  }
}